// DistributedMemory_54348516164186
// MI455X (gfx1250) — compile-verified
//
#include <hip/hip_runtime.h>

typedef __attribute__((ext_vector_type(2))) float v2f;
typedef __attribute__((ext_vector_type(8))) float v8f;

#define VEC_DIM    128
#define CTX        8
#define N_SAMPLES  6
#define TILE_B     16
#define WAVES_PER_BLOCK 4
#define ROW_PAD    4
#define ROW_STRIDE (VEC_DIM + ROW_PAD)   // 132 floats: rows land on distinct LDS banks

// One wave owns one tile of 16 batch rows. Phase 1 stages inputs[16][128] into LDS
// with fully-coalesced 512B row gathers (float4 per lane). Phase 2 runs, per sample
// slot s, a 16x16x128 f32 GEMM via 32 chained V_WMMA_F32_16X16X4_F32 and extracts
// the diagonal. NW > 0 specializes the outputs-matrix column count at compile time
// so the strided column gathers become base + 32-bit-constant offsets (no serial
// 64-bit address chains / xcnt stalls).
template <int NW>
__global__ __launch_bounds__(WAVES_PER_BLOCK * 32)
void pvdm_logits_wmma_kernel(const int* __restrict__ doc_ids,
                             const int* __restrict__ context_ids,  // [B][CTX]
                             const int* __restrict__ sample_ids,   // [B][N_SAMPLES]
                             const float* __restrict__ paragraph,  // [N_DOCS][128]
                             const float* __restrict__ words,      // [N_WORDS][128]
                             const float* __restrict__ outputs,    // [128][n_words]
                             float* __restrict__ logits,           // [B][N_SAMPLES]
                             int batch, int n_words_rt)
{
    const int nw = (NW > 0) ? NW : n_words_rt;

    __shared__ float lds[WAVES_PER_BLOCK * TILE_B * ROW_STRIDE];

    const int lane = threadIdx.x & 31;
    const int wave = threadIdx.x >> 5;
    const int tile = blockIdx.x * WAVES_PER_BLOCK + wave;
    const int b0   = tile * TILE_B;
    if (b0 >= batch) return;            // wave-uniform: EXEC stays all-ones below

    float* tileLds = &lds[wave * (TILE_B * ROW_STRIDE)];

    // ---------------- Phase 1: inputs tile -> LDS (coalesced row gathers) ----
    for (int r = 0; r < TILE_B; ++r) {
        int b = b0 + r;
        if (b >= batch) b = batch - 1;                    // clamp (no EXEC change)
        const float4* drow =
            (const float4*)(paragraph + (size_t)doc_ids[b] * VEC_DIM);
        float4 acc = drow[lane];                          // 32 lanes x 16B = one 512B row
#pragma unroll
        for (int c = 0; c < CTX; ++c) {
            const float4* wrow =
                (const float4*)(words + (size_t)context_ids[b * CTX + c] * VEC_DIM);
            float4 w = wrow[lane];
            acc.x += w.x; acc.y += w.y; acc.z += w.z; acc.w += w.w;
        }
        *(float4*)(tileLds + r * ROW_STRIDE + lane * 4) = acc;   // ds_store_b128
    }
    // Same-wave cross-lane LDS visibility: DS ops are in-order per wave; fence
    // the compiler and wait the CDNA5 split DS counter explicitly.
    asm volatile("s_wait_dscnt 0" ::: "memory");

    // ---------------- Phase 2: D = A(16x128) x B(128x16) per sample slot -----
    // A fragment (16x4 f32, 2 VGPRs/step): lane<16 -> M=lane, K=4k+{0,1};
    //                                      lane>=16 -> M=lane-16, K=4k+{2,3}.
    const int rowSel = lane & 15;
    const int delta  = (lane >> 4) * 2;
    const float* aBase = tileLds + rowSel * ROW_STRIDE + delta;

    int bRow = b0 + rowSel;
    if (bRow >= batch) bRow = batch - 1;

    for (int s = 0; s < N_SAMPLES; ++s) {
        const int col = sample_ids[bRow * N_SAMPLES + s];
        // This lane's B column element index for K-row (delta + d):
        //   outputs[(delta + d) * nw + col]  -- 32-bit element offsets; with NW
        // compile-time the per-k terms are constants folded into the addressing.
        const uint32_t colD = (uint32_t)col + (uint32_t)delta * (uint32_t)nw;

        v8f acc = {0.f, 0.f, 0.f, 0.f, 0.f, 0.f, 0.f, 0.f};
#pragma unroll
        for (int k = 0; k < VEC_DIM / 4; ++k) {
            float2 ap = *(const float2*)(aBase + 4 * k);  // ds_load_b64
            v2f a; a[0] = ap.x; a[1] = ap.y;
            v2f bb;
            bb[0] = outputs[colD + (uint32_t)(4 * k + 0) * (uint32_t)nw];
            bb[1] = outputs[colD + (uint32_t)(4 * k + 1) * (uint32_t)nw];
            acc = __builtin_amdgcn_wmma_f32_16x16x4_f32(
                false, a, false, bb, (short)0, acc, false, false);
        }

        // Diagonal extraction: j<8 -> lane j holds acc[j];
        //                      j>=8 -> lane j+16 holds acc[j-8].
        const int sel = lane & 7;
        float v = acc[0];
        if (sel == 1) v = acc[1];
        if (sel == 2) v = acc[2];
        if (sel == 3) v = acc[3];
        if (sel == 4) v = acc[4];
        if (sel == 5) v = acc[5];
        if (sel == 6) v = acc[6];
        if (sel == 7) v = acc[7];

        const bool active = (lane < 8) || (lane >= 24);
        const int  j      = (lane < 8) ? lane : (lane - 16);
        const int  bOut   = b0 + j;
        if (active && bOut < batch)
            logits[bOut * N_SAMPLES + s] = v;
    }
}

extern "C" void kernel_launch(void* const* d_in, const int* in_sizes, int n_in,
                              void* d_out, int out_size, void* d_ws, size_t ws_size,
                              hipStream_t stream) {
    const int*   doc_ids     = (const int*)d_in[0];
    const int*   context_ids = (const int*)d_in[1];
    const int*   sample_ids  = (const int*)d_in[2];
    const float* paragraph   = (const float*)d_in[3];
    const float* words       = (const float*)d_in[4];
    const float* outputs     = (const float*)d_in[5];
    float*       logits      = (float*)d_out;

    const int batch   = in_sizes[0];                 // 16384
    const int n_words = in_sizes[5] / VEC_DIM;       // outputs is [128][n_words]

    const int tiles  = (batch + TILE_B - 1) / TILE_B;
    const int blocks = (tiles + WAVES_PER_BLOCK - 1) / WAVES_PER_BLOCK;

    if (n_words == 100000) {
        pvdm_logits_wmma_kernel<100000>
            <<<blocks, WAVES_PER_BLOCK * 32, 0, stream>>>(
                doc_ids, context_ids, sample_ids, paragraph, words, outputs,
                logits, batch, n_words);
    } else {
        pvdm_logits_wmma_kernel<0>
            <<<blocks, WAVES_PER_BLOCK * 32, 0, stream>>>(
                doc_ids, context_ids, sample_ids, paragraph, words, outputs,
                logits, batch, n_words);
    }
}